// GroupedQueryAttention_91001767067684
// MI455X (gfx1250) — compile-verified
//
#include <hip/hip_runtime.h>
#include <hip/hip_bf16.h>
#include <math.h>

// ---------------------------------------------------------------------------
// GQA forward on MI455X (gfx1250), wave32, all matmuls via V_WMMA_F32_16X16X4_F32.
// Ping-pong double buffering (no register copies) + immediate-offset addressing.
// ---------------------------------------------------------------------------

typedef __attribute__((ext_vector_type(2))) float f32x2;
typedef __attribute__((ext_vector_type(8))) float f32x8;

#define WMMA_F32X4(a, b, c) \
    __builtin_amdgcn_wmma_f32_16x16x4_f32(false, (a), false, (b), (short)0, (c), false, false)

static constexpr int kB   = 2;
static constexpr int kS   = 2048;
static constexpr int kD   = 1024;
static constexpr int kNH  = 16;
static constexpr int kNKV = 4;
static constexpr int kDK  = 64;

__device__ __forceinline__ f32x8 zero8() {
    f32x8 z = {0.f, 0.f, 0.f, 0.f, 0.f, 0.f, 0.f, 0.f};
    return z;
}

// ---------------------------------------------------------------------------
// f32 GEMM: C[M,N] = A[M,K] @ B[K,N], row-major. N,K compile-time constants.
// Block = 128 threads = 4 waves; each wave owns a 64x64 tile (4x4 WMMA tiles).
// Steady-state loop is 2x unrolled with ping-pong frag buffers: buffer X is
// reloaded (for step +2) immediately after its WMMAs issue — no copies, and
// s_wait_loadcnt only waits past the in-flight batch.
// ---------------------------------------------------------------------------
template <int N, int K>
__global__ __launch_bounds__(128) void gemm_f32_wmma(
    const float* __restrict__ A, const float* __restrict__ B, float* __restrict__ C,
    int M)
{
    const int lane = threadIdx.x & 31;
    const int wv   = threadIdx.x >> 5;     // 0..3
    const int l15  = lane & 15;
    const int kl   = (lane >> 4) * 2;      // K sub-slot: 0 (lanes 0-15) / 2 (lanes 16-31)

    const int bm = blockIdx.y * 128 + (wv >> 1) * 64;
    const int bn = blockIdx.x * 128 + (wv & 1) * 64;

    f32x8 acc[4][4];
    #pragma unroll
    for (int i = 0; i < 4; ++i)
        #pragma unroll
        for (int j = 0; j < 4; ++j)
            acc[i][j] = zero8();

    const float* Arow[4];
    #pragma unroll
    for (int i = 0; i < 4; ++i)
        Arow[i] = A + (size_t)(bm + i * 16 + l15) * K + kl;
    const float* Bcol = B + (size_t)kl * N + bn + l15;

    auto loadA = [&](int k, f32x2* a) {
        #pragma unroll
        for (int i = 0; i < 4; ++i)
            a[i] = *reinterpret_cast<const f32x2*>(Arow[i] + k);
    };
    auto loadB = [&](int k, f32x2* bfr) {
        const float* bp = Bcol + (size_t)k * N;
        #pragma unroll
        for (int j = 0; j < 4; ++j) {
            bfr[j].x = bp[j * 16];
            bfr[j].y = bp[j * 16 + N];
        }
    };
    auto mm16 = [&](const f32x2* a, const f32x2* bfr) {
        #pragma unroll
        for (int i = 0; i < 4; ++i)
            #pragma unroll
            for (int j = 0; j < 4; ++j)
                acc[i][j] = WMMA_F32X4(a[i], bfr[j], acc[i][j]);
    };

    f32x2 a0[4], b0[4], a1[4], b1[4];
    loadA(0, a0); loadB(0, b0);
    loadA(4, a1); loadB(4, b1);

    #pragma unroll 1
    for (int k = 0; k + 8 < K; k += 8) {
        if ((k & 31) == 0) {            // warm WGP$ 64 K-steps ahead
            #pragma unroll
            for (int i = 0; i < 4; ++i)
                __builtin_prefetch(Arow[i] + k + 64, 0, 0);
            __builtin_prefetch(Bcol + (size_t)(k + 64) * N, 0, 0);
            __builtin_prefetch(Bcol + (size_t)(k + 64) * N + 48, 0, 0);
        }
        mm16(a0, b0);
        loadA(k + 8, a0); loadB(k + 8, b0);       // refill buffer 0 for step +2
        mm16(a1, b1);
        loadA(k + 12, a1); loadB(k + 12, b1);     // refill buffer 1 for step +2
    }
    mm16(a0, b0);
    mm16(a1, b1);

    // ---- store (C/D layout: vgpr r -> row r / r+8 per half-wave) ----
    const int rlo = (lane >> 4) * 8;
    #pragma unroll
    for (int i = 0; i < 4; ++i) {
        float* crow = C + (size_t)(bm + i * 16 + rlo) * N + bn + l15;
        #pragma unroll
        for (int j = 0; j < 4; ++j)
            #pragma unroll
            for (int r = 0; r < 8; ++r)
                crow[(size_t)r * N + j * 16] = acc[i][j][r];
    }
}

// ---------------------------------------------------------------------------
// RoPE in-place: buf shaped [(b*S + s)*nheads + h][64]; pairs (i, i+32), half=32.
// ---------------------------------------------------------------------------
__global__ void rope_kernel(float* __restrict__ buf, int nheads, int total)
{
    int idx = blockIdx.x * blockDim.x + threadIdx.x;
    if (idx >= total) return;
    int i   = idx & 31;
    int rem = idx >> 5;                       // (b*S + s)*nheads + h
    int bs  = rem / nheads;                   // b*S + s
    int s   = bs & (kS - 1);

    // inv_freq = 10000^(-i/32) = exp(-i * ln(10000)/32)
    float inv_freq = expf(-(float)i * (9.210340371976184f / 32.0f));
    float ang = (float)s * inv_freq;
    float c = cosf(ang);
    float sn = sinf(ang);

    float* base = buf + (size_t)rem * 64;
    float x1 = base[i];
    float x2 = base[i + 32];
    base[i]      = x1 * c - x2 * sn;
    base[i + 32] = x1 * sn + x2 * c;
}

// ---------------------------------------------------------------------------
// Flash-attention: one wave per (qtile of 16 rows, head, batch). d_k=64.
//  q,o: [(b*S+s)*16 + h][64]   k,v: [(b*S+s)*4 + kv][64]
// Ping-pong K-tile buffers (no copies); all K/V loads are immediate offsets
// from two advancing base pointers (one constant 64-bit add per block).
// ---------------------------------------------------------------------------
__global__ __launch_bounds__(32) void attn_kernel(
    const float* __restrict__ qb, const float* __restrict__ kb,
    const float* __restrict__ vb, float* __restrict__ ob)
{
    __shared__ float plds[16 * 18];           // padded stride 18: 8B-aligned, conflict-free

    const int lane = threadIdx.x & 31;
    const int l15  = lane & 15;
    const int hi   = lane >> 4;               // half-wave: 0/1
    const int kl   = hi * 2;
    const int qt   = blockIdx.x;              // 0..127  (query tile)
    const int h    = blockIdx.y;              // 0..15   (q head)
    const int b    = blockIdx.z;              // 0..1
    const int kv   = h >> 2;                  // KV head (GROUP=4)
    const int qbase = qt * 16;

    constexpr int ROW = kNKV * kDK;           // 256 floats between consecutive keys
    constexpr int BS  = 16 * ROW;             // floats per 16-key block

    // Preload Q A-frags: lane l15 = query row, frag s covers feature cols 4s..4s+3
    const float* qrow = qb + (((size_t)b * kS + qbase + l15) * kNH + h) * kDK;
    f32x2 qf[16];
    #pragma unroll
    for (int s = 0; s < 16; ++s)
        qf[s] = *reinterpret_cast<const f32x2*>(qrow + 4 * s + kl);

    f32x8 o[4];
    #pragma unroll
    for (int t = 0; t < 4; ++t) o[t] = zero8();
    float mrow[8], lrow[8];
    #pragma unroll
    for (int r = 0; r < 8; ++r) { mrow[r] = -3.0e38f; lrow[r] = 0.f; }

    const float scale = 0.125f;               // 1/sqrt(64)

    // Per-lane base pointers for block 0 (advanced by BS per key block).
    const float* kblk = kb + (((size_t)b * kS + l15) * kNKV + kv) * kDK + kl;
    const float* vblk = vb + (((size_t)b * kS + kl) * kNKV + kv) * kDK + l15;

    // K-tile: B-frag layout (lane l15 = key row, contiguous float2 in d)
    auto loadK = [&](const float* kp, f32x2* kf) {
        #pragma unroll
        for (int s = 0; s < 16; ++s)
            kf[s] = *reinterpret_cast<const f32x2*>(kp + 4 * s);
    };

    // Process one 16-key block; vp = V base for this block, all offsets immediate.
    auto process = [&](const float* vp, const f32x2* kf, bool diag) {
        // ---- S = Q @ K^T ----
        f32x8 s8 = zero8();
        #pragma unroll
        for (int s = 0; s < 16; ++s)
            s8 = WMMA_F32X4(qf[s], kf[s], s8);

        // ---- issue V loads now; consumed only after the softmax below ----
        f32x2 vf[4][4];
        #pragma unroll
        for (int s2 = 0; s2 < 4; ++s2)
            #pragma unroll
            for (int t = 0; t < 4; ++t) {
                vf[s2][t].x = vp[s2 * 4 * ROW + t * 16];
                vf[s2][t].y = vp[s2 * 4 * ROW + ROW + t * 16];
            }

        // ---- scale + causal mask + online softmax ----
        #pragma unroll
        for (int r = 0; r < 8; ++r) {
            float v = s8[r] * scale;
            if (diag && (l15 > r + hi * 8)) v = -1.0e30f;   // key > query

            float rm = v;
            rm = fmaxf(rm, __shfl_xor(rm, 1, 32));
            rm = fmaxf(rm, __shfl_xor(rm, 2, 32));
            rm = fmaxf(rm, __shfl_xor(rm, 4, 32));
            rm = fmaxf(rm, __shfl_xor(rm, 8, 32));

            float mnew = fmaxf(mrow[r], rm);
            float corr = __expf(mrow[r] - mnew);
            float p    = __expf(v - mnew);

            float rs = p;
            rs += __shfl_xor(rs, 1, 32);
            rs += __shfl_xor(rs, 2, 32);
            rs += __shfl_xor(rs, 4, 32);
            rs += __shfl_xor(rs, 8, 32);

            lrow[r] = lrow[r] * corr + rs;
            mrow[r] = mnew;
            #pragma unroll
            for (int t = 0; t < 4; ++t) o[t][r] *= corr;

            plds[(r + hi * 8) * 18 + l15] = p;   // C/D-layout row -> LDS
        }
        __syncthreads();   // single-wave WG: DS ordering; barrier is NOP

        // ---- transpose P (C/D layout -> A layout) via LDS ----
        f32x2 pf[4];
        #pragma unroll
        for (int s2 = 0; s2 < 4; ++s2)
            pf[s2] = *reinterpret_cast<const f32x2*>(&plds[l15 * 18 + 4 * s2 + kl]);

        // ---- O += P @ V ----
        #pragma unroll
        for (int s2 = 0; s2 < 4; ++s2)
            #pragma unroll
            for (int t = 0; t < 4; ++t)
                o[t] = WMMA_F32X4(pf[s2], vf[s2][t], o[t]);
        __syncthreads();
    };

    // ---- pipelined key-block loop: ping-pong buffers, diagonal block peeled ----
    f32x2 k0[16], k1[16];
    loadK(kblk, k0);
    int j = 0;
    #pragma unroll 1
    for (; j + 2 <= qt; j += 2) {
        loadK(kblk + BS, k1);               // in flight during process(j)
        process(vblk, k0, false);
        loadK(kblk + 2 * BS, k0);           // in flight during process(j+1)
        process(vblk + BS, k1, false);
        kblk += 2 * BS;
        vblk += 2 * BS;
    }
    if (j < qt) {                           // one non-diagonal block remains
        loadK(kblk + BS, k1);
        process(vblk, k0, false);
        process(vblk + BS, k1, true);
    } else {
        process(vblk, k0, true);
    }

    // ---- normalize + store ----
    #pragma unroll
    for (int r = 0; r < 8; ++r) {
        float inv = 1.0f / lrow[r];
        int row = qbase + r + hi * 8;
        float* orow = ob + (((size_t)b * kS + row) * kNH + h) * kDK;
        #pragma unroll
        for (int t = 0; t < 4; ++t)
            orow[t * 16 + l15] = o[t][r] * inv;
    }
}

// ---------------------------------------------------------------------------
extern "C" void kernel_launch(void* const* d_in, const int* in_sizes, int n_in,
                              void* d_out, int out_size, void* d_ws, size_t ws_size,
                              hipStream_t stream)
{
    const float* x  = (const float*)d_in[0];
    const float* wq = (const float*)d_in[1];
    const float* wk = (const float*)d_in[2];
    const float* wv = (const float*)d_in[3];
    const float* wo = (const float*)d_in[4];
    float* out = (float*)d_out;

    const int M = kB * kS;                 // 4096
    float* qbuf = (float*)d_ws;            // 4096*1024
    float* kbuf = qbuf + (size_t)M * kD;   // 4096*256
    float* vbuf = kbuf + (size_t)M * (kNKV * kDK);
    float* abuf = vbuf + (size_t)M * (kNKV * kDK);   // 4096*1024

    dim3 blk(128);

    // Projections
    gemm_f32_wmma<kD, kD><<<dim3(kD / 128, M / 128), blk, 0, stream>>>(x, wq, qbuf, M);
    gemm_f32_wmma<kNKV * kDK, kD><<<dim3((kNKV * kDK) / 128, M / 128), blk, 0, stream>>>(x, wk, kbuf, M);
    gemm_f32_wmma<kNKV * kDK, kD><<<dim3((kNKV * kDK) / 128, M / 128), blk, 0, stream>>>(x, wv, vbuf, M);

    // RoPE on q and k
    {
        int tq = M * kNH * 32;
        rope_kernel<<<(tq + 255) / 256, 256, 0, stream>>>(qbuf, kNH, tq);
        int tk = M * kNKV * 32;
        rope_kernel<<<(tk + 255) / 256, 256, 0, stream>>>(kbuf, kNKV, tk);
    }

    // Flash attention: one wave per (qtile, head, batch)
    attn_kernel<<<dim3(kS / 16, kNH, kB), dim3(32), 0, stream>>>(qbuf, kbuf, vbuf, abuf);

    // Output projection
    gemm_f32_wmma<kD, kD><<<dim3(kD / 128, M / 128), blk, 0, stream>>>(abuf, wo, out, M);
}